// Sparsemax_26018911879367
// MI455X (gfx1250) — compile-verified
//
#include <hip/hip_runtime.h>
#include <hip/hip_bf16.h>

#define D_DIM   4096
#define BLOCK   256
#define WAVES   (BLOCK / 32)
#define PT      (D_DIM / BLOCK)      // 16 elements per thread
#define VPT     (PT / 4)             // 4 float4 per thread
#define BISECT  30

typedef int v4i __attribute__((ext_vector_type(4)));
typedef __attribute__((address_space(1))) v4i   gbl_v4i_t;
typedef __attribute__((address_space(3))) v4i   lds_v4i_t;
typedef __attribute__((address_space(3))) float lds_f32_t;

__device__ __forceinline__ float wave_sum(float v) {
#pragma unroll
  for (int off = 16; off > 0; off >>= 1) v += __shfl_xor(v, off, 32);
  return v;
}

__device__ __forceinline__ float wave_max(float v) {
#pragma unroll
  for (int off = 16; off > 0; off >>= 1) v = fmaxf(v, __shfl_xor(v, off, 32));
  return v;
}

// Block-wide sum, broadcast to all threads. 8 wave32 partials combined in LDS.
__device__ __forceinline__ float block_sum(float v, float* s_red, float* s_bc) {
  const int lane = threadIdx.x & 31;
  const int wv   = threadIdx.x >> 5;
  v = wave_sum(v);
  __syncthreads();                       // protect s_red/s_bc from previous use
  if (lane == 0) s_red[wv] = v;
  __syncthreads();
  if (threadIdx.x == 0) {
    float t = s_red[0];
#pragma unroll
    for (int i = 1; i < WAVES; ++i) t += s_red[i];
    *s_bc = t;
  }
  __syncthreads();
  return *s_bc;
}

__global__ __launch_bounds__(BLOCK) void sparsemax_kernel(const float* __restrict__ x,
                                                          float* __restrict__ out) {
  __shared__ float s_row[D_DIM];         // row data, later holds z = x - max
  __shared__ float s_cand[D_DIM];        // compacted candidates (z > -1)
  __shared__ float s_red[WAVES];
  __shared__ float s_bc;
  __shared__ int   s_wcnt[WAVES];

  const int tid  = threadIdx.x;
  const int lane = tid & 31;
  const int wv   = tid >> 5;
  const size_t row = blockIdx.x;

  const float4* __restrict__ xrow4 = (const float4*)(x + row * (size_t)D_DIM);
  float4* __restrict__ orow4       = (float4*)(out + row * (size_t)D_DIM);

  // ---- Stage 1: async DMA the 16 KB row global -> LDS (ASYNCcnt path) ----
#pragma unroll
  for (int c = 0; c < VPT; ++c) {
    const int j = c * BLOCK + tid;       // float4 index; lanes hit consecutive 16B
#if __has_builtin(__builtin_amdgcn_global_load_async_to_lds_b128)
    __builtin_amdgcn_global_load_async_to_lds_b128(
        (gbl_v4i_t*)(xrow4 + j),
        (lds_v4i_t*)(lds_f32_t*)&s_row[4 * j], 0, 0);
#else
    {
      unsigned lds_off = (unsigned)(size_t)(lds_f32_t*)&s_row[4 * j];
      unsigned long long ga = (unsigned long long)(xrow4 + j);
      asm volatile("global_load_async_to_lds_b128 %0, %1, off"
                   :: "v"(lds_off), "v"(ga) : "memory");
    }
#endif
  }
#if __has_builtin(__builtin_amdgcn_s_wait_asynccnt)
  __builtin_amdgcn_s_wait_asynccnt(0);
#else
  asm volatile("s_wait_asynccnt 0" ::: "memory");
#endif
  __syncthreads();

  // ---- Stage 2: row max (block reduce) ----
  float m = -3.402823466e38f;
#pragma unroll
  for (int c = 0; c < PT; ++c) m = fmaxf(m, s_row[tid + c * BLOCK]);
  m = wave_max(m);
  if (lane == 0) s_red[wv] = m;
  __syncthreads();
  if (tid == 0) {
    float t = s_red[0];
#pragma unroll
    for (int i = 1; i < WAVES; ++i) t = fmaxf(t, s_red[i]);
    s_bc = t;
  }
  __syncthreads();
  m = s_bc;

  // ---- Stage 3: z = x - m; deterministic compaction of candidates z > -1 ----
  // (tau* >= -1, so z <= -1 can never be in the support nor contribute to f)
  float zv[PT];
  int cnt = 0;
#pragma unroll
  for (int c = 0; c < PT; ++c) {
    const int idx = tid + c * BLOCK;
    const float z = s_row[idx] - m;
    zv[c] = z;
    s_row[idx] = z;
    cnt += (z > -1.0f) ? 1 : 0;
  }
  // wave-inclusive scan of per-thread counts
  int inc = cnt;
#pragma unroll
  for (int off = 1; off < 32; off <<= 1) {
    int n = __shfl_up(inc, off, 32);
    if (lane >= off) inc += n;
  }
  if (lane == 31) s_wcnt[wv] = inc;
  __syncthreads();
  int base = 0, ncand = 0;
#pragma unroll
  for (int i = 0; i < WAVES; ++i) {
    const int t = s_wcnt[i];
    if (i < wv) base += t;
    ncand += t;
  }
  int p = base + (inc - cnt);            // exclusive offset for this thread
#pragma unroll
  for (int c = 0; c < PT; ++c)
    if (zv[c] > -1.0f) s_cand[p++] = zv[c];
  __syncthreads();

  // ---- Stage 4: bisection on f(tau) = sum relu(c - tau) - 1 over [-1, 0),
  //               then 2 Michelot fixed-point corrections for exact tau ----
  float tau;
  if (ncand <= 32) {
    // Common case (~14 candidates): single wave32, shuffle-only, no barriers.
    if (tid < 32) {
      const float c = (tid < ncand) ? s_cand[tid] : -2.0f;  // filler never contributes
      float lo = -1.0f, hi = 0.0f;
      for (int it = 0; it < BISECT; ++it) {
        const float mid = 0.5f * (lo + hi);
        const float f = wave_sum(fmaxf(c - mid, 0.0f)) - 1.0f;
        if (f >= 0.0f) lo = mid; else hi = mid;
      }
      float t = lo;                      // t < 0 always, so z_max = 0 keeps k >= 1
      for (int r = 0; r < 2; ++r) {
        const float s = wave_sum((c > t) ? c : 0.0f);
        const float k = wave_sum((c > t) ? 1.0f : 0.0f);
        t = (s - 1.0f) / k;
      }
      if (tid == 0) s_bc = t;
    }
    __syncthreads();
    tau = s_bc;
  } else {
    float lo = -1.0f, hi = 0.0f;
    for (int it = 0; it < BISECT; ++it) {
      const float mid = 0.5f * (lo + hi);
      float part = 0.0f;
      for (int i = tid; i < ncand; i += BLOCK) part += fmaxf(s_cand[i] - mid, 0.0f);
      const float f = block_sum(part, s_red, &s_bc) - 1.0f;
      if (f >= 0.0f) lo = mid; else hi = mid;
    }
    float t = lo;
    for (int r = 0; r < 2; ++r) {
      float ps = 0.0f, pk = 0.0f;
      for (int i = tid; i < ncand; i += BLOCK) {
        const float c = s_cand[i];
        if (c > t) { ps += c; pk += 1.0f; }
      }
      const float s = block_sum(ps, s_red, &s_bc);
      const float k = block_sum(pk, s_red, &s_bc);
      t = (s - 1.0f) / k;
    }
    tau = t;
  }

  // ---- Stage 5: out = max(z - tau, 0), vectorized B128 stores ----
#pragma unroll
  for (int c = 0; c < VPT; ++c) {
    const int j = c * BLOCK + tid;
    const float4 v = ((const float4*)s_row)[j];
    float4 r;
    r.x = fmaxf(v.x - tau, 0.0f);
    r.y = fmaxf(v.y - tau, 0.0f);
    r.z = fmaxf(v.z - tau, 0.0f);
    r.w = fmaxf(v.w - tau, 0.0f);
    orow4[j] = r;
  }
}

extern "C" void kernel_launch(void* const* d_in, const int* in_sizes, int n_in,
                              void* d_out, int out_size, void* d_ws, size_t ws_size,
                              hipStream_t stream) {
  const float* x = (const float*)d_in[0];
  float* out = (float*)d_out;
  const int rows = in_sizes[0] / D_DIM;   // 16384 x 4096
  sparsemax_kernel<<<dim3(rows), dim3(BLOCK), 0, stream>>>(x, out);
  (void)n_in; (void)out_size; (void)d_ws; (void)ws_size;
}